// DQATTEN_75831942578202
// MI455X (gfx1250) — compile-verified
//
#include <hip/hip_runtime.h>

// CDNA5 / gfx1250 fused kernel for the DQ-attention reference.
//
// Strategy (memory-bound after refactor; ~404 MiB traffic => ~18us floor):
//   scores[h,n] = obs[n,:] . (q[h,:] @ Wk[h])   -- avoids materializing k
// Two WMMA f32 16x16x4 GEMMs per 16-row tile:
//   QH[16,160] = St[16,256] @ [Wq^T | Sb_w1^T]  (ReLU epilogues)
//   QK[16,128] per head = Qh[16,32] @ Wk[h]
// Everything else (z mean, scores, softmax, weighted z-sum) is VALU with
// coalesced float4/float2 streams and wave32 butterfly reductions.

typedef float v2f __attribute__((ext_vector_type(2)));
typedef float v8f __attribute__((ext_vector_type(8)));

#define BT       16384
#define N_       32
#define NQ_      64
#define S_       256
#define O_       128
#define H_       4
#define E_       32
#define ROWS     16
#define THREADS  256
#define MASKV    (-999999.0f)

// LDS row strides (floats), padded so WMMA-A reads (16 lanes, same k,
// different row) hit distinct banks.
#define ST_STR   260   // states tile  [16][256]
#define QH_STR   164   // q/h acts     [16][160]
#define QK_STR   516   // qk           [16][4*128]

__global__ __launch_bounds__(THREADS)
void dqatten_fused(const float* __restrict__ z,      // [BT][32][64]
                   const float* __restrict__ states, // [BT][256]
                   const float* __restrict__ obs,    // [BT][32][128]
                   const float* __restrict__ Wq,     // [4][32][256]  (H,E,S)
                   const float* __restrict__ Wk,     // [4][32][128]  (H,E,O)
                   const float* __restrict__ Sb_w1,  // [32][256]
                   const float* __restrict__ Sb_b1,  // [32]
                   const float* __restrict__ Sb_w2,  // [32]
                   const float* __restrict__ Sb_b2,  // [1]
                   float* __restrict__ out)          // [BT][64]
{
  __shared__ float St[ROWS * ST_STR];
  __shared__ float QH[ROWS * QH_STR];
  __shared__ float QK[ROWS * QK_STR];
  __shared__ float QV[ROWS * N_];
  __shared__ float SC[ROWS * N_ * H_];
  __shared__ float WL[ROWS * N_];
  __shared__ float VB[ROWS];

  const int tid  = threadIdx.x;
  const int wave = tid >> 5;        // 8 waves (wave32)
  const int lane = tid & 31;
  const int hi   = lane >> 4;       // 0: lanes 0-15, 1: lanes 16-31
  const int m    = lane & 15;
  const int r0   = blockIdx.x * ROWS;

  // ---------------- Phase A: stage states, prefetch obs, z-mean ----------
  {
    // 16 contiguous state rows (4096 f32), fully coalesced.
    const float* src = states + (size_t)r0 * S_;
    for (int i = tid; i < ROWS * S_; i += THREADS) {
      int r = i >> 8, k = i & 255;
      St[r * ST_STR + k] = src[i];
    }
    // Warm L2/WGP$ with this WG's obs block (256 KiB) -> global_prefetch_b8.
    const char* ob = (const char*)(obs + (size_t)r0 * N_ * O_);
    for (int i = tid; i < 1024; i += THREADS)
      __builtin_prefetch(ob + (size_t)i * 256, 0, 1);

    // q_vals[r][n] = mean_nq z : 4-lane groups, 64B-contiguous float4 loads.
    const int grp = lane >> 2, j = lane & 3;
    for (int it = 0; it < 8; ++it) {
      int p = wave * 64 + it * 8 + grp;             // 0..511 = (r,n)
      int r = p >> 5, n = p & 31;
      const float* zp = z + ((size_t)(r0 + r) * N_ + n) * NQ_ + j * 4;
      float s = 0.f;
      #pragma unroll
      for (int c = 0; c < 4; ++c) {
        float4 v4 = *(const float4*)(zp + c * 16);
        s += v4.x + v4.y + v4.z + v4.w;
      }
      s += __shfl_xor(s, 1, 32);
      s += __shfl_xor(s, 2, 32);
      if (j == 0) QV[r * N_ + n] = s * (1.0f / NQ_);
    }
  }
  __syncthreads();

  // ---------------- Phase B: QH = ReLU(St @ [Wq^T | Sb_w1^T]) (WMMA) -----
  {
    const float inv_sqrt_e = 0.17677669529663687f;  // 1/sqrt(E), folded here
    for (int tile = wave; tile < 10; tile += 8) {   // wave-uniform loop
      const int col0 = tile * 16;
      const float* Wt = (tile < 8) ? (Wq + (size_t)col0 * S_)
                                   : (Sb_w1 + (size_t)(col0 - 128) * S_);
      v8f acc = {};
      for (int kk = 0; kk < S_; kk += 4) {
        const int k = kk + 2 * hi;
        // A 16x4 f32: lane m holds row m, VGPR0/1 = K {k,k+1} (hi=0) / {k+2,k+3}
        v2f a = *(const v2f*)&St[m * ST_STR + k];
        // B 4x16: lane m holds column col0+m; memory is col-major-in-k (W[c][k])
        v2f b = *(const v2f*)&Wt[(size_t)m * S_ + k];
        acc = __builtin_amdgcn_wmma_f32_16x16x4_f32(false, a, false, b,
                                                    (short)0, acc, false, false);
      }
      const int col = col0 + m;
      if (tile < 8) {                 // q columns: ReLU, fold 1/sqrt(E)
        #pragma unroll
        for (int r = 0; r < 8; ++r)
          QH[(r + 8 * hi) * QH_STR + col] = fmaxf(acc[r], 0.f) * inv_sqrt_e;
      } else {                        // MLP hidden: +bias, ReLU
        const float b1 = Sb_b1[col - 128];
        #pragma unroll
        for (int r = 0; r < 8; ++r)
          QH[(r + 8 * hi) * QH_STR + col] = fmaxf(acc[r] + b1, 0.f);
      }
    }
  }
  __syncthreads();

  // ---------------- Phase C: v[r] scalar MLP out + QK GEMM (WMMA) --------
  {
    const float w2 = Sb_w2[lane];
    const float b2 = Sb_b2[0];
    for (int rr = wave; rr < ROWS; rr += 8) {
      float s = w2 * QH[rr * QH_STR + 128 + lane];
      s += __shfl_xor(s, 16, 32);
      s += __shfl_xor(s, 8, 32);
      s += __shfl_xor(s, 4, 32);
      s += __shfl_xor(s, 2, 32);
      s += __shfl_xor(s, 1, 32);
      if (lane == 0) VB[rr] = s + b2;
    }
    // qk[r][h][o] = sum_e QH[r][h*32+e] * Wk[h][e][o] : per-head 16x128x32
    for (int job = wave; job < 32; job += 8) {      // wave-uniform
      const int h = job >> 3, o0 = (job & 7) * 16;
      const float* wk = Wk + (size_t)h * E_ * O_;
      v8f acc = {};
      for (int kk = 0; kk < E_; kk += 4) {
        const int k = kk + 2 * hi;
        v2f a = *(const v2f*)&QH[m * QH_STR + h * E_ + k];
        v2f b;
        b.x = wk[(size_t)k * O_ + o0 + m];
        b.y = wk[(size_t)(k + 1) * O_ + o0 + m];
        acc = __builtin_amdgcn_wmma_f32_16x16x4_f32(false, a, false, b,
                                                    (short)0, acc, false, false);
      }
      #pragma unroll
      for (int r = 0; r < 8; ++r)
        QK[(r + 8 * hi) * QK_STR + h * O_ + o0 + m] = acc[r];
    }
  }
  __syncthreads();

  // ---------------- Phase D: scores, mask, softmax, head-sum weights -----
  {
    for (int rr = wave * 2; rr < wave * 2 + 2; ++rr) {
      const size_t g = (size_t)(r0 + rr);
      // Hoist qk for this row: lane owns o = 4*lane..4*lane+3 per head.
      float4 qk0 = *(const float4*)&QK[rr * QK_STR + 0 * O_ + 4 * lane];
      float4 qk1 = *(const float4*)&QK[rr * QK_STR + 1 * O_ + 4 * lane];
      float4 qk2 = *(const float4*)&QK[rr * QK_STR + 2 * O_ + 4 * lane];
      float4 qk3 = *(const float4*)&QK[rr * QK_STR + 3 * O_ + 4 * lane];
      for (int n = 0; n < N_; ++n) {
        const float4 ob = *(const float4*)&obs[(g * N_ + n) * O_ + 4 * lane];
        float s0 = ob.x*qk0.x + ob.y*qk0.y + ob.z*qk0.z + ob.w*qk0.w;
        float s1 = ob.x*qk1.x + ob.y*qk1.y + ob.z*qk1.z + ob.w*qk1.w;
        float s2 = ob.x*qk2.x + ob.y*qk2.y + ob.z*qk2.z + ob.w*qk2.w;
        float s3 = ob.x*qk3.x + ob.y*qk3.y + ob.z*qk3.z + ob.w*qk3.w;
        #pragma unroll
        for (int off = 16; off; off >>= 1) {
          s0 += __shfl_xor(s0, off, 32);
          s1 += __shfl_xor(s1, off, 32);
          s2 += __shfl_xor(s2, off, 32);
          s3 += __shfl_xor(s3, off, 32);
        }
        if (lane == 0) {
          const bool msk = (QV[rr * N_ + n] <= MASKV);
          float* sp = &SC[(rr * N_ + n) * H_];
          sp[0] = msk ? MASKV : s0;
          sp[1] = msk ? MASKV : s1;
          sp[2] = msk ? MASKV : s2;
          sp[3] = msk ? MASKV : s3;
        }
      }
      // softmax over n (lane == n), per head; then sum heads.
      {
        const float* sp = &SC[(rr * N_ + lane) * H_];
        float s0 = sp[0], s1 = sp[1], s2 = sp[2], s3 = sp[3];
        float m0 = s0, m1 = s1, m2 = s2, m3 = s3;
        #pragma unroll
        for (int off = 16; off; off >>= 1) {
          m0 = fmaxf(m0, __shfl_xor(m0, off, 32));
          m1 = fmaxf(m1, __shfl_xor(m1, off, 32));
          m2 = fmaxf(m2, __shfl_xor(m2, off, 32));
          m3 = fmaxf(m3, __shfl_xor(m3, off, 32));
        }
        float e0 = __expf(s0 - m0), e1 = __expf(s1 - m1);
        float e2 = __expf(s2 - m2), e3 = __expf(s3 - m3);
        float t0 = e0, t1 = e1, t2 = e2, t3 = e3;
        #pragma unroll
        for (int off = 16; off; off >>= 1) {
          t0 += __shfl_xor(t0, off, 32);
          t1 += __shfl_xor(t1, off, 32);
          t2 += __shfl_xor(t2, off, 32);
          t3 += __shfl_xor(t3, off, 32);
        }
        WL[rr * N_ + lane] = e0 / t0 + e1 / t1 + e2 / t2 + e3 / t3 + 1e-10f;
      }
    }
  }
  __syncthreads();

  // ---------------- Phase E: out[nq] = sum_n w[n]*z[n,nq] + 32*v ---------
  {
    for (int rr = wave * 2; rr < wave * 2 + 2; ++rr) {
      const size_t g = (size_t)(r0 + rr);
      const float bias = (float)N_ * VB[rr];      // sum_n of broadcast v
      float2 acc = make_float2(bias, bias);
      const float* zr = z + g * N_ * NQ_;
      for (int n = 0; n < N_; ++n) {
        const float  wn = WL[rr * N_ + n];        // LDS broadcast
        const float2 z2 = *(const float2*)(zr + n * NQ_ + 2 * lane);
        acc.x += wn * z2.x;
        acc.y += wn * z2.y;
      }
      *(float2*)&out[g * NQ_ + 2 * lane] = acc;
    }
  }
}

extern "C" void kernel_launch(void* const* d_in, const int* in_sizes, int n_in,
                              void* d_out, int out_size, void* d_ws, size_t ws_size,
                              hipStream_t stream) {
  (void)in_sizes; (void)n_in; (void)out_size; (void)d_ws; (void)ws_size;
  dqatten_fused<<<BT / ROWS, THREADS, 0, stream>>>(
      (const float*)d_in[0],   // z_values
      (const float*)d_in[1],   // states
      (const float*)d_in[2],   // obs
      (const float*)d_in[3],   // Wq
      (const float*)d_in[4],   // Wk
      (const float*)d_in[5],   // Sb_w1
      (const float*)d_in[6],   // Sb_b1
      (const float*)d_in[7],   // Sb_w2
      (const float*)d_in[8],   // Sb_b2
      (float*)d_out);
}